// PackedAbsMeanScaler_22832046146263
// MI455X (gfx1250) — compile-verified
//
#include <hip/hip_runtime.h>
#include <hip/hip_bf16.h>

typedef __attribute__((ext_vector_type(2))) float v2f;
typedef __attribute__((ext_vector_type(8))) float v8f;

#define B_      16
#define L_      8192
#define D_      256
#define NVARS   16
#define NGROUPS 128
#define MINSCALE 1e-5f

#define CHUNK 64   // L rows staged per iteration
#define DTILE 16   // feature columns per workgroup

// ---------------------------------------------------------------------------
// Kernel 1: segment reduction as one-hot GEMM on V_WMMA_F32_16X16X4_F32.
//   grid = (D/16, B), block = 256 threads = 8 waves; wave m owns groups
//   [16m, 16m+16). Target/mask staged once through LDS, shared by all 8 waves.
//   Chunks with no key in a wave's group tile are skipped (wave-uniform
//   branch, EXEC stays all-ones as WMMA requires).
// ---------------------------------------------------------------------------
__global__ __launch_bounds__(256)
void seg_reduce_wmma(const float* __restrict__ target,
                     const unsigned char* __restrict__ mask,
                     const int* __restrict__ sample_id,
                     const int* __restrict__ variate_id,
                     float* __restrict__ gscale)
{
    __shared__ int   s_keys[CHUNK];
    __shared__ float s_msk[CHUNK][DTILE];   // mask as f32
    __shared__ float s_abs[CHUNK][DTILE];   // |t| * mask

    const int tid   = threadIdx.x;
    const int lane  = tid & 31;
    const int wave  = tid >> 5;            // 8 waves -> 8 group tiles
    const int b     = blockIdx.y;
    const int dbase = blockIdx.x * DTILE;
    const int gbase = wave * 16;

    // ISA 16x4 f32 A layout: lanes 0-15 hold M=0..15 with K={0,1} in v0/v1,
    // lanes 16-31 hold K={2,3}. B (4x16) assumed mirrored with N across lanes.
    const int row  = lane & 15;            // A row (group), B/C column (d)
    const int half = lane >> 4;
    const int kb   = half * 2;

    v8f c_cnt = {0.f,0.f,0.f,0.f,0.f,0.f,0.f,0.f};
    v8f c_sum = {0.f,0.f,0.f,0.f,0.f,0.f,0.f,0.f};

    for (int l0 = 0; l0 < L_; l0 += CHUNK) {
        // ---- stage keys -------------------------------------------------
        if (tid < CHUNK) {
            int l = l0 + tid;
            s_keys[tid] = sample_id[b * L_ + l] * NVARS + variate_id[b * L_ + l];
        }
        // ---- stage 64x16 tile of mask / |t|*mask ------------------------
        #pragma unroll
        for (int i = 0; i < (CHUNK * DTILE) / 256; ++i) {
            int idx = tid + 256 * i;
            int lo  = idx >> 4;
            int n   = idx & 15;
            size_t ga = (size_t)(b * L_ + l0 + lo) * D_ + dbase + n;
            float t = target[ga];
            float m = mask[ga] ? 1.0f : 0.0f;
            s_msk[lo][n] = m;
            s_abs[lo][n] = fabsf(t) * m;
            if (l0 + CHUNK < L_)  // prefetch next chunk (global_prefetch_b8)
                __builtin_prefetch(&target[ga + (size_t)CHUNK * D_], 0, 3);
        }
        __syncthreads();

        // ---- wave-uniform sparsity skip ---------------------------------
        int ka = s_keys[lane];
        int kc2 = s_keys[lane + 32];
        bool hit = ((unsigned)(ka  - gbase) < 16u) ||
                   ((unsigned)(kc2 - gbase) < 16u);
        if (__builtin_amdgcn_ballot_w32(hit)) {
            #pragma unroll
            for (int kc = 0; kc < CHUNK / 4; ++kc) {
                int kk = kc * 4 + kb;
                int key0 = s_keys[kk];
                int key1 = s_keys[kk + 1];
                v2f a, bm, ba;
                a.x  = (key0 == gbase + row) ? 1.0f : 0.0f;   // one-hot A
                a.y  = (key1 == gbase + row) ? 1.0f : 0.0f;
                bm.x = s_msk[kk][row];     bm.y = s_msk[kk + 1][row];
                ba.x = s_abs[kk][row];     ba.y = s_abs[kk + 1][row];
                c_cnt = __builtin_amdgcn_wmma_f32_16x16x4_f32(
                            false, a, false, bm, (short)0, c_cnt, false, false);
                c_sum = __builtin_amdgcn_wmma_f32_16x16x4_f32(
                            false, a, false, ba, (short)0, c_sum, false, false);
            }
        }
        __syncthreads();
    }

    // ---- finalize: gs = max(count? asum/count : 0, 1e-5) ----------------
    // 16x16 f32 C layout: VGPR r holds M = r + 8*half, N = row.
    #pragma unroll
    for (int r = 0; r < 8; ++r) {
        int g = gbase + r + half * 8;
        int d = dbase + row;
        float cnt = c_cnt[r];
        float sum = c_sum[r];
        float gs  = (cnt > 0.0f) ? (sum / cnt) : 0.0f;
        gs = fmaxf(gs, MINSCALE);
        gscale[((size_t)b * NGROUPS + g) * D_ + d] = gs;
    }
}

// ---------------------------------------------------------------------------
// Kernel 2: pure-bandwidth broadcast. float4 over B*L*D; writes loc=0 and
// scale = gscale[b, key, d] (or 1.0 where sample_id==0).
// ---------------------------------------------------------------------------
__global__ __launch_bounds__(256)
void broadcast_scale(const int* __restrict__ sample_id,
                     const int* __restrict__ variate_id,
                     const float4* __restrict__ gscale4,
                     float4* __restrict__ loc4,
                     float4* __restrict__ scale4)
{
    size_t idx = (size_t)blockIdx.x * 256 + threadIdx.x;  // float4 index
    int    d4  = (int)(idx & (D_ / 4 - 1));               // 64 float4 per row
    size_t bl  = idx >> 6;                                // b*L + l
    int sid = sample_id[bl];
    int vid = variate_id[bl];
    int b   = (int)(bl >> 13);                            // L = 2^13
    int key = sid * NVARS + vid;
    float4 gs = gscale4[((size_t)b * NGROUPS + key) * (D_ / 4) + d4];
    if (sid == 0) gs = make_float4(1.f, 1.f, 1.f, 1.f);
    loc4[idx]   = make_float4(0.f, 0.f, 0.f, 0.f);
    scale4[idx] = gs;
}

// ---------------------------------------------------------------------------
extern "C" void kernel_launch(void* const* d_in, const int* in_sizes, int n_in,
                              void* d_out, int out_size, void* d_ws, size_t ws_size,
                              hipStream_t stream)
{
    const float*         target = (const float*)d_in[0];
    const unsigned char* mask   = (const unsigned char*)d_in[1]; // jax bool = 1B
    const int*           sid    = (const int*)d_in[2];
    const int*           vid    = (const int*)d_in[3];
    float*               out    = (float*)d_out;
    float*               gscale = (float*)d_ws;  // B*G*D*4 = 2 MB scratch

    dim3 grid1(D_ / DTILE, B_);                  // 16 x 16 = 256 workgroups
    seg_reduce_wmma<<<grid1, 256, 0, stream>>>(target, mask, sid, vid, gscale);

    size_t nd = (size_t)B_ * L_ * D_;            // elements per output tensor
    int blocks2 = (int)(nd / 4 / 256);           // float4 granularity
    broadcast_scale<<<blocks2, 256, 0, stream>>>(
        sid, vid, (const float4*)gscale,
        (float4*)out,            // loc
        (float4*)(out + nd));    // scale
}